// GINEncoder_31963146617270
// MI455X (gfx1250) — compile-verified
//
#include <hip/hip_runtime.h>
#include <hip/hip_bf16.h>

// CDNA5 / gfx1250, wave32. WMMA f32 16x16x4: A,B = 2 VGPRs/lane (v2f),
// C/D = 8 VGPRs/lane (v8f).
typedef float v2f __attribute__((ext_vector_type(2)));
typedef float v8f __attribute__((ext_vector_type(8)));

#define WAVES_PER_BLOCK 8          // 256 threads = 8 wave32
#define NODES_PER_BLOCK (WAVES_PER_BLOCK * 16)

// Guaranteed-native f32 scatter-add: GLOBAL_ATOMIC_ADD_F32, no-return form
// (STOREcnt path), device scope so the RMW resolves at the device-wide L2.
// Avoids the backend's conservative CAS-loop expansion of atomicrmw fadd.
__device__ __forceinline__ void atomic_add_f32(float* p, float v) {
    asm volatile("global_atomic_add_f32 %0, %1, off scope:SCOPE_DEV"
                 :: "v"(p), "v"(v)
                 : "memory");
}

// ---------------------------------------------------------------- zero init
__global__ void k_zero(float* __restrict__ p, int n) {
    int i = blockIdx.x * blockDim.x + threadIdx.x;
    int stride = gridDim.x * blockDim.x;
    for (; i < n; i += stride) p[i] = 0.0f;
}

// ------------------------------------------------- scatter pass 1: agg1 += x[src]
__global__ void k_scatter1(const long long* __restrict__ src,
                           const long long* __restrict__ dst,
                           const float* __restrict__ x,
                           float* __restrict__ agg1, int n_edges) {
    int e = blockIdx.x * blockDim.x + threadIdx.x;
    if (e >= n_edges) return;
    int s = (int)src[e];
    int d = (int)dst[e];
    float4 v = ((const float4*)x)[s];           // x is (N,4) -> one float4 per node
    float* p = agg1 + (size_t)d * 4;
    atomic_add_f32(p + 0, v.x);
    atomic_add_f32(p + 1, v.y);
    atomic_add_f32(p + 2, v.z);
    atomic_add_f32(p + 3, v.w);
}

// ------------------------------------------------- scatter pass 2: agg2 += h[src]
// One thread per (edge, 4-feature chunk): 4 threads/edge, 16 feats total.
__global__ void k_scatter2(const long long* __restrict__ src,
                           const long long* __restrict__ dst,
                           const float* __restrict__ h,
                           float* __restrict__ agg2, int n_edges) {
    int tid = blockIdx.x * blockDim.x + threadIdx.x;
    int e = tid >> 2;
    if (e >= n_edges) return;
    int q = tid & 3;
    int s = (int)src[e];
    int d = (int)dst[e];
    float4 v = ((const float4*)h)[(size_t)s * 4 + q];   // h is (N,16)
    float* p = agg2 + (size_t)d * 16 + q * 4;
    atomic_add_f32(p + 0, v.x);
    atomic_add_f32(p + 1, v.y);
    atomic_add_f32(p + 2, v.z);
    atomic_add_f32(p + 3, v.w);
}

// ------------------------------------------------- MLP1 (4->16 relu ->16) + ELU
// One wave handles a tile of 16 nodes. Layer1: single V_WMMA_F32_16X16X4_F32
// (K=4 exactly). Layer2: K=16 as 4 chained K=4 WMMAs with C accumulation.
// A layout (16x4 f32): lanes 0-15 hold M=0..15 / K={0,1}; lanes 16-31 K={2,3}.
// C/D layout: VGPR r: lanes 0-15 -> M=r, lanes 16-31 -> M=r+8, N=lane%16.
__global__ void k_mlp1(const float* __restrict__ x,
                       const float* __restrict__ agg1,
                       const float* __restrict__ W1, const float* __restrict__ b1,
                       const float* __restrict__ W2, const float* __restrict__ b2,
                       float* __restrict__ h, int n_nodes) {
    __shared__ float tile[WAVES_PER_BLOCK][16][17];   // pad 17: no LDS bank conflicts

    const int lane = threadIdx.x & 31;
    const int wave = threadIdx.x >> 5;
    const int half = lane >> 4;      // 0: K 0..1 | 1: K 2..3
    const int m16  = lane & 15;      // node row (A) / column N (B,C,D)
    const int tile_base = (blockIdx.x * WAVES_PER_BLOCK + wave) * 16;
    const bool valid = (tile_base < n_nodes);   // wave-uniform (N%16==0 -> full tiles)

    if (valid) {
        const int node = tile_base + m16;
        // A = x + agg1 (GIN eps=0 aggregate), 16x4
        v2f a;
        a.x = x[(size_t)node * 4 + 2 * half + 0] + agg1[(size_t)node * 4 + 2 * half + 0];
        a.y = x[(size_t)node * 4 + 2 * half + 1] + agg1[(size_t)node * 4 + 2 * half + 1];
        // B = W1 (4x16 row-major): lane n holds column n, K per half
        v2f b;
        b.x = W1[(2 * half + 0) * 16 + m16];
        b.y = W1[(2 * half + 1) * 16 + m16];
        // C = bias broadcast down each column
        v8f c;
        const float bias = b1[m16];
        #pragma unroll
        for (int r = 0; r < 8; ++r) c[r] = bias;

        c = __builtin_amdgcn_wmma_f32_16x16x4_f32(false, a, false, b,
                                                  (short)0, c, false, false);
        // ReLU, stash 16x16 tile in LDS in (m,n) order for A-layout reload
        #pragma unroll
        for (int r = 0; r < 8; ++r) {
            float v = c[r] > 0.0f ? c[r] : 0.0f;
            tile[wave][r + 8 * half][m16] = v;
        }
    }
    __syncthreads();
    if (valid) {
        v8f c;
        const float bias2 = b2[m16];
        #pragma unroll
        for (int r = 0; r < 8; ++r) c[r] = bias2;
        // Layer 2: K=16 -> 4 chained 16x16x4 WMMAs
        #pragma unroll
        for (int kc = 0; kc < 4; ++kc) {
            const int k0 = 4 * kc + 2 * half;
            v2f a, b;
            a.x = tile[wave][m16][k0 + 0];
            a.y = tile[wave][m16][k0 + 1];
            b.x = W2[(k0 + 0) * 16 + m16];
            b.y = W2[(k0 + 1) * 16 + m16];
            c = __builtin_amdgcn_wmma_f32_16x16x4_f32(false, a, false, b,
                                                      (short)0, c, false, false);
        }
        // ELU between GIN layers, write h (N,16)
        #pragma unroll
        for (int r = 0; r < 8; ++r) {
            float v = c[r];
            v = v > 0.0f ? v : (__expf(v) - 1.0f);
            h[(size_t)(tile_base + r + 8 * half) * 16 + m16] = v;
        }
    }
}

// ------------------------------------------------- MLP2: relu((h+agg2) @ W3 + b3)
// K=16 -> 4 chained K=4 WMMAs; N=8 (columns 8..15 masked to zero on load/store).
__global__ void k_mlp2(const float* __restrict__ h,
                       const float* __restrict__ agg2,
                       const float* __restrict__ W3, const float* __restrict__ b3,
                       float* __restrict__ out, int n_nodes) {
    const int lane = threadIdx.x & 31;
    const int wave = threadIdx.x >> 5;
    const int half = lane >> 4;
    const int m16  = lane & 15;
    const int tile_base = (blockIdx.x * WAVES_PER_BLOCK + wave) * 16;
    if (tile_base >= n_nodes) return;   // wave-uniform; no barrier in this kernel

    const bool ncol = (m16 < 8);
    v8f c;
    const float bias = ncol ? b3[m16] : 0.0f;
    #pragma unroll
    for (int r = 0; r < 8; ++r) c[r] = bias;

    const int node = tile_base + m16;
    #pragma unroll
    for (int kc = 0; kc < 4; ++kc) {
        const int k0 = 4 * kc + 2 * half;
        v2f a, b;
        a.x = h[(size_t)node * 16 + k0 + 0] + agg2[(size_t)node * 16 + k0 + 0];
        a.y = h[(size_t)node * 16 + k0 + 1] + agg2[(size_t)node * 16 + k0 + 1];
        b.x = ncol ? W3[(k0 + 0) * 8 + m16] : 0.0f;
        b.y = ncol ? W3[(k0 + 1) * 8 + m16] : 0.0f;
        c = __builtin_amdgcn_wmma_f32_16x16x4_f32(false, a, false, b,
                                                  (short)0, c, false, false);
    }
    if (ncol) {
        #pragma unroll
        for (int r = 0; r < 8; ++r) {
            float v = c[r] > 0.0f ? c[r] : 0.0f;
            out[(size_t)(tile_base + r + 8 * half) * 8 + m16] = v;
        }
    }
}

// ---------------------------------------------------------------- launch
extern "C" void kernel_launch(void* const* d_in, const int* in_sizes, int n_in,
                              void* d_out, int out_size, void* d_ws, size_t ws_size,
                              hipStream_t stream) {
    const float*     x  = (const float*)d_in[0];
    const long long* ei = (const long long*)d_in[1];   // int64 (2, E)
    const float* W1 = (const float*)d_in[2];
    const float* b1 = (const float*)d_in[3];
    const float* W2 = (const float*)d_in[4];
    const float* b2 = (const float*)d_in[5];
    const float* W3 = (const float*)d_in[6];
    const float* b3 = (const float*)d_in[7];
    float* out = (float*)d_out;

    const int n_nodes = in_sizes[0] / 4;
    const int n_edges = in_sizes[1] / 2;
    const long long* src = ei;
    const long long* dst = ei + n_edges;

    // workspace: agg1 (N*4) | agg2 (N*16) | h (N*16)
    float* agg1 = (float*)d_ws;
    float* agg2 = agg1 + (size_t)n_nodes * 4;
    float* hbuf = agg2 + (size_t)n_nodes * 16;

    const int zn = n_nodes * 20;                 // agg1+agg2 contiguous
    k_zero<<<1024, 256, 0, stream>>>(agg1, zn);

    k_scatter1<<<(n_edges + 255) / 256, 256, 0, stream>>>(src, dst, x, agg1, n_edges);

    const int mlp_grid = (n_nodes + NODES_PER_BLOCK - 1) / NODES_PER_BLOCK;
    k_mlp1<<<mlp_grid, 32 * WAVES_PER_BLOCK, 0, stream>>>(x, agg1, W1, b1, W2, b2,
                                                          hbuf, n_nodes);

    const long long s2_threads = (long long)n_edges * 4;
    k_scatter2<<<(int)((s2_threads + 255) / 256), 256, 0, stream>>>(src, dst, hbuf,
                                                                    agg2, n_edges);

    k_mlp2<<<mlp_grid, 32 * WAVES_PER_BLOCK, 0, stream>>>(hbuf, agg2, W3, b3,
                                                          out, n_nodes);
}